// GNNRoutingModel_463856468120
// MI455X (gfx1250) — compile-verified
//
#include <hip/hip_runtime.h>

typedef __attribute__((ext_vector_type(2))) float v2f;
typedef __attribute__((ext_vector_type(8))) float v8f;

constexpr int N_NODES = 1024;
constexpr int D_INF   = 128;
constexpr int NH      = 4;     // heads
constexpr int HC      = 32;    // per-head dim
constexpr int F_OUT   = 128;   // NH*HC
constexpr int E_EDGES = 16384;
constexpr int NE      = E_EDGES + N_NODES;  // + self loops

// ---------------------------------------------------------------------------
// C[M,N] = A[M,K] @ B[K,N], row-major. One wave (32 lanes) per 16x16 C tile.
// Uses V_WMMA_F32_16X16X4_F32 (full fp32 precision, CDNA5 matrix pipe).
// A 16x4 tile: lanes 0-15 = rows M, VGPR0/1 = K 0/1 (lanes 0-15), K 2/3 (16-31).
// B 4x16 tile: lanes hold N columns, same K split across lane halves.
// C/D: VGPR r -> M=r (lanes 0-15), M=r+8 (lanes 16-31); N = lane&15.
// ---------------------------------------------------------------------------
__global__ void wmma_gemm_f32_kernel(const float* __restrict__ A,
                                     const float* __restrict__ B,
                                     float* __restrict__ Cm,
                                     int K, int lda, int ldb, int ldc) {
  const int lane = threadIdx.x;        // 0..31, no divergence: EXEC all ones
  const int half = lane >> 4;          // 0 or 1
  const int lid  = lane & 15;
  const int m0 = blockIdx.y * 16;
  const int n0 = blockIdx.x * 16;

  v8f acc = {};
  const float* arow = A + (size_t)(m0 + lid) * lda;
  const float* bcol = B + n0 + lid;

  for (int kk = 0; kk < K; kk += 4) {
    v2f a, b;
    a.x = arow[kk + 2 * half];
    a.y = arow[kk + 2 * half + 1];
    b.x = bcol[(size_t)(kk + 2 * half) * ldb];
    b.y = bcol[(size_t)(kk + 2 * half + 1) * ldb];
    acc = __builtin_amdgcn_wmma_f32_16x16x4_f32(false, a, false, b,
                                                (short)0, acc, false, false);
  }
#pragma unroll
  for (int r = 0; r < 8; ++r) {
    Cm[(size_t)(m0 + r + 8 * half) * ldc + n0 + lid] = acc[r];
  }
}

// ---------------------------------------------------------------------------
// a_src[n,h] = sum_c h[n,h,c]*att_s[h,c];  a_dst likewise.
// ---------------------------------------------------------------------------
__global__ void att_coef_kernel(const float* __restrict__ h,
                                const float* __restrict__ att_s,
                                const float* __restrict__ att_d,
                                float* __restrict__ a_src,
                                float* __restrict__ a_dst) {
  int idx = blockIdx.x * blockDim.x + threadIdx.x;   // n*NH + hh
  if (idx >= N_NODES * NH) return;
  int n = idx >> 2, hh = idx & 3;
  const float* hv = h + (size_t)n * F_OUT + hh * HC;
  const float* as = att_s + hh * HC;
  const float* ad = att_d + hh * HC;
  float ss = 0.f, sd = 0.f;
#pragma unroll
  for (int c = 0; c < HC; ++c) { ss += hv[c] * as[c]; sd += hv[c] * ad[c]; }
  a_src[idx] = ss;
  a_dst[idx] = sd;
}

__device__ __forceinline__ void edge_sd(const int* __restrict__ ei, int e,
                                        int& s, int& d) {
  if (e < E_EDGES) { s = ei[e]; d = ei[E_EDGES + e]; }
  else             { s = d = e - E_EDGES; }          // self loop
}

__device__ __forceinline__ unsigned enc_f32(float x) {
  unsigned u = __float_as_uint(x);
  return (u & 0x80000000u) ? ~u : (u | 0x80000000u);
}
__device__ __forceinline__ float dec_f32(unsigned u) {
  return (u & 0x80000000u) ? __uint_as_float(u & 0x7FFFFFFFu)
                           : __uint_as_float(~u);
}

// pass 1: e = leaky_relu(a_src[src]+a_dst[dst], 0.2); segment max via atomicMax
__global__ void edge_logits_kernel(const int* __restrict__ ei,
                                   const float* __restrict__ a_src,
                                   const float* __restrict__ a_dst,
                                   float* __restrict__ e_buf,
                                   unsigned* __restrict__ m_bits) {
  int idx = blockIdx.x * blockDim.x + threadIdx.x;   // e*NH + hh
  if (idx >= NE * NH) return;
  int e = idx >> 2, hh = idx & 3;
  int s, d; edge_sd(ei, e, s, d);
  float v = a_src[s * NH + hh] + a_dst[d * NH + hh];
  v = v > 0.f ? v : 0.2f * v;
  e_buf[idx] = v;
  atomicMax(&m_bits[d * NH + hh], enc_f32(v));
}

// pass 2: e = exp(e - m[dst]); segment sum of denominators
__global__ void edge_exp_kernel(const int* __restrict__ ei,
                                const unsigned* __restrict__ m_bits,
                                float* __restrict__ e_buf,
                                float* __restrict__ s_sum) {
  int idx = blockIdx.x * blockDim.x + threadIdx.x;
  if (idx >= NE * NH) return;
  int e = idx >> 2, hh = idx & 3;
  int s, d; edge_sd(ei, e, s, d);
  float m = dec_f32(m_bits[d * NH + hh]);
  float v = __expf(e_buf[idx] - m);
  e_buf[idx] = v;
  atomicAdd(&s_sum[d * NH + hh], v);
}

// pass 3: agg[dst,f] += alpha * h[src,f]
__global__ void edge_scatter_kernel(const int* __restrict__ ei,
                                    const float* __restrict__ e_buf,
                                    const float* __restrict__ s_sum,
                                    const float* __restrict__ h,
                                    float* __restrict__ agg) {
  int gid = blockIdx.x * blockDim.x + threadIdx.x;   // e*F_OUT + f
  if (gid >= NE * F_OUT) return;
  int e = gid >> 7, f = gid & 127, hh = f >> 5;
  int s, d; edge_sd(ei, e, s, d);
  float alpha = e_buf[e * NH + hh] / (s_sum[d * NH + hh] + 1e-16f);
  atomicAdd(&agg[(size_t)d * F_OUT + f], alpha * h[(size_t)s * F_OUT + f]);
}

__global__ void bias_lrelu_kernel(float* __restrict__ x,
                                  const float* __restrict__ b, float slope) {
  int idx = blockIdx.x * blockDim.x + threadIdx.x;
  if (idx >= N_NODES * F_OUT) return;
  float v = x[idx] + b[idx & (F_OUT - 1)];
  x[idx] = v > 0.f ? v : slope * v;
}

// out[i,j] = sum_k relu(p_row[i,k]+p_col[j,k]+fc1_b[k]) * fc2_w[k] + fc2_b
__global__ void pairwise_kernel(const float* __restrict__ p_row,
                                const float* __restrict__ p_col,
                                const float* __restrict__ fc1_b,
                                const float* __restrict__ fc2_w,
                                const float* __restrict__ fc2_b,
                                float* __restrict__ out) {
  __shared__ float rowt[16][HC];
  __shared__ float colt[16][HC];
  __shared__ float bsh[HC];
  __shared__ float wsh[HC];
  int tid = threadIdx.x;                 // 256 threads
  int i0 = blockIdx.y * 16, j0 = blockIdx.x * 16;
  for (int t = tid; t < 16 * HC; t += 256) {
    int r = t >> 5, c = t & 31;
    rowt[r][c] = p_row[(size_t)(i0 + r) * HC + c];
    colt[r][c] = p_col[(size_t)(j0 + r) * HC + c];
  }
  if (tid < HC) { bsh[tid] = fc1_b[tid]; wsh[tid] = fc2_w[tid]; }
  __syncthreads();
  int il = tid >> 4, jl = tid & 15;
  float acc = fc2_b[0];
#pragma unroll
  for (int k = 0; k < HC; ++k) {
    float z = rowt[il][k] + colt[jl][k] + bsh[k];
    acc += fmaxf(z, 0.f) * wsh[k];
  }
  out[(size_t)(i0 + il) * N_NODES + j0 + jl] = acc;
}

// ---------------------------------------------------------------------------
static void run_gat_layer(const float* in, int K, const float* W,
                          const float* as_, const float* ad_, const float* b,
                          const int* ei, float* hbuf, float* agg,
                          float* a_src, float* a_dst, unsigned* m_bits,
                          float* s_sum, float* e_buf, hipStream_t stream) {
  dim3 gg(F_OUT / 16, N_NODES / 16);
  wmma_gemm_f32_kernel<<<gg, 32, 0, stream>>>(in, W, hbuf, K, K, F_OUT, F_OUT);
  att_coef_kernel<<<(N_NODES * NH + 127) / 128, 128, 0, stream>>>(
      hbuf, as_, ad_, a_src, a_dst);
  hipMemsetAsync(m_bits, 0, N_NODES * NH * sizeof(unsigned), stream);
  hipMemsetAsync(s_sum, 0, N_NODES * NH * sizeof(float), stream);
  hipMemsetAsync(agg, 0, (size_t)N_NODES * F_OUT * sizeof(float), stream);
  edge_logits_kernel<<<(NE * NH + 255) / 256, 256, 0, stream>>>(
      ei, a_src, a_dst, e_buf, m_bits);
  edge_exp_kernel<<<(NE * NH + 255) / 256, 256, 0, stream>>>(
      ei, m_bits, e_buf, s_sum);
  edge_scatter_kernel<<<(NE * F_OUT + 255) / 256, 256, 0, stream>>>(
      ei, e_buf, s_sum, hbuf, agg);
  bias_lrelu_kernel<<<(N_NODES * F_OUT + 255) / 256, 256, 0, stream>>>(
      agg, b, 0.01f);
}

extern "C" void kernel_launch(void* const* d_in, const int* in_sizes, int n_in,
                              void* d_out, int out_size, void* d_ws,
                              size_t ws_size, hipStream_t stream) {
  (void)in_sizes; (void)n_in; (void)out_size; (void)ws_size;
  const float* x    = (const float*)d_in[0];
  const int*   ei   = (const int*)d_in[1];
  const float* W1   = (const float*)d_in[2];
  const float* a1s  = (const float*)d_in[3];
  const float* a1d  = (const float*)d_in[4];
  const float* b1   = (const float*)d_in[5];
  const float* W2   = (const float*)d_in[6];
  const float* a2s  = (const float*)d_in[7];
  const float* a2d  = (const float*)d_in[8];
  const float* b2   = (const float*)d_in[9];
  const float* W3   = (const float*)d_in[10];
  const float* a3s  = (const float*)d_in[11];
  const float* a3d  = (const float*)d_in[12];
  const float* b3   = (const float*)d_in[13];
  const float* fc1w = (const float*)d_in[14];
  const float* fc1b = (const float*)d_in[15];
  const float* fc2w = (const float*)d_in[16];
  const float* fc2b = (const float*)d_in[17];

  float* ws = (float*)d_ws;
  float* bufH   = ws;                                 // [1024,128] h = x@W
  float* bufP   = bufH + N_NODES * F_OUT;             // [1024,128] ping
  float* bufQ   = bufP + N_NODES * F_OUT;             // [1024,128] pong
  float* a_src  = bufQ + N_NODES * F_OUT;             // [1024,4]
  float* a_dst  = a_src + N_NODES * NH;               // [1024,4]
  unsigned* m_bits = (unsigned*)(a_dst + N_NODES * NH);  // [1024,4]
  float* s_sum  = (float*)(m_bits + N_NODES * NH);    // [1024,4]
  float* e_buf  = s_sum + N_NODES * NH;               // [17408,4]
  float* p_row  = e_buf + NE * NH;                    // [1024,32]
  float* p_col  = p_row + N_NODES * HC;               // [1024,32]

  // three GAT layers (ping-pong agg buffers; layer output = agg after bias+lrelu)
  run_gat_layer(x,    D_INF, W1, a1s, a1d, b1, ei, bufH, bufP,
                a_src, a_dst, m_bits, s_sum, e_buf, stream);
  run_gat_layer(bufP, F_OUT, W2, a2s, a2d, b2, ei, bufH, bufQ,
                a_src, a_dst, m_bits, s_sum, e_buf, stream);
  run_gat_layer(bufQ, F_OUT, W3, a3s, a3d, b3, ei, bufH, bufP,
                a_src, a_dst, m_bits, s_sum, e_buf, stream);

  // p_row = h @ fc1_w[:128], p_col = h @ fc1_w[128:]  (h = bufP)
  dim3 gp(HC / 16, N_NODES / 16);
  wmma_gemm_f32_kernel<<<gp, 32, 0, stream>>>(bufP, fc1w, p_row,
                                              F_OUT, F_OUT, HC, HC);
  wmma_gemm_f32_kernel<<<gp, 32, 0, stream>>>(bufP, fc1w + F_OUT * HC, p_col,
                                              F_OUT, F_OUT, HC, HC);

  // out[i,j] = relu-dot over 32 channels
  dim3 go(N_NODES / 16, N_NODES / 16);
  pairwise_kernel<<<go, 256, 0, stream>>>(p_row, p_col, fc1b, fc2w, fc2b,
                                          (float*)d_out);
}